// BlockLinear_85864986182075
// MI455X (gfx1250) — compile-verified
//
#include <hip/hip_runtime.h>

typedef float v2f __attribute__((ext_vector_type(2)));
typedef float v8f __attribute__((ext_vector_type(8)));
typedef int   v4i __attribute__((ext_vector_type(4)));

#define BATCH   1024
#define NBLK    1024
#define DIN     64
#define DOUT    64
#define TILE_M  128
#define LDSW    68   // 64 + 4 pad: 272B row stride -> 16B aligned, bank-spread

// global (addrspace 1) and LDS (addrspace 3) v4i pointers for the async builtin
#define GPTR(p) ((__attribute__((address_space(1))) v4i*)(p))
#define LPTR(p) ((__attribute__((address_space(3))) v4i*)(p))

#define HAS_ASYNC_LDS __has_builtin(__builtin_amdgcn_global_load_async_to_lds_b128)

__global__ __launch_bounds__(256, 1)
void block_linear_wmma_f32(const float* __restrict__ x,
                           const float* __restrict__ W,
                           const float* __restrict__ bias,
                           float* __restrict__ out)
{
    __shared__ float Xs[TILE_M * LDSW];   // 34816 B (reused as output staging)
    __shared__ float Ws[DOUT   * LDSW];   // 17408 B
    __shared__ float Bs[DOUT];            //   256 B

    const int n   = blockIdx.x;              // diagonal block index
    const int m0  = blockIdx.y * TILE_M;     // batch-row base
    const int tid = threadIdx.x;             // 0..255

    if (tid < DOUT) Bs[tid] = bias[n * DOUT + tid];

    // ---------------- tile loads: global -> LDS -----------------------------
#if HAS_ASYNC_LDS
    // CDNA5 async copy: bypasses VGPRs, tracked by ASYNCcnt.
#pragma unroll
    for (int t = 0; t < 8; ++t) {            // X tile: 128 rows x 16 x b128
        int j = tid + t * 256, row = j >> 4, col4 = j & 15;
        const float* src = x + (size_t)((m0 + row) * NBLK + n) * DIN + col4 * 4;
        __builtin_amdgcn_global_load_async_to_lds_b128(
            GPTR(src), LPTR(&Xs[row * LDSW + col4 * 4]), 0, 0);
    }
#pragma unroll
    for (int t = 0; t < 4; ++t) {            // W tile: 64 rows x 16 x b128
        int j = tid + t * 256, o = j >> 4, col4 = j & 15;
        const float* src = W + (size_t)(n * DOUT + o) * DIN + col4 * 4;
        __builtin_amdgcn_global_load_async_to_lds_b128(
            GPTR(src), LPTR(&Ws[o * LDSW + col4 * 4]), 0, 0);
    }
#if __has_builtin(__builtin_amdgcn_s_wait_asynccnt)
    __builtin_amdgcn_s_wait_asynccnt(0);
#else
    asm volatile("s_wait_asynccnt 0" ::: "memory");
#endif
#else
    // Fallback: global -> VGPR -> LDS
    {
        const float4* xg = reinterpret_cast<const float4*>(x);
#pragma unroll
        for (int t = 0; t < 8; ++t) {
            int j = tid + t * 256, row = j >> 4, col4 = j & 15;
            float4 v = xg[(size_t)((m0 + row) * NBLK + n) * (DIN / 4) + col4];
            *reinterpret_cast<float4*>(&Xs[row * LDSW + col4 * 4]) = v;
        }
        const float4* wg = reinterpret_cast<const float4*>(W);
#pragma unroll
        for (int t = 0; t < 4; ++t) {
            int j = tid + t * 256, o = j >> 4, col4 = j & 15;
            float4 v = wg[(size_t)(n * DOUT + o) * (DIN / 4) + col4];
            *reinterpret_cast<float4*>(&Ws[o * LDSW + col4 * 4]) = v;
        }
    }
#endif
    __syncthreads();

    // ---------------- WMMA compute: each wave = 16-row strip x 64 DOUT ------
    const int lane  = tid & 31;
    const int wave  = tid >> 5;      // 0..7
    const int half  = lane >> 4;     // lanes 16-31 hold K+2 halves
    const int lrow  = lane & 15;     // M for A-frag, N for B/C/D frags
    const int rbase = wave * 16;

    v8f acc0 = {}, acc1 = {}, acc2 = {}, acc3 = {};

    const float* xr = &Xs[(rbase + lrow) * LDSW + 2 * half];
    const float* w0 = &Ws[( 0 + lrow) * LDSW + 2 * half];
    const float* w1 = &Ws[(16 + lrow) * LDSW + 2 * half];
    const float* w2 = &Ws[(32 + lrow) * LDSW + 2 * half];
    const float* w3 = &Ws[(48 + lrow) * LDSW + 2 * half];

#pragma unroll
    for (int k = 0; k < DIN; k += 4) {
        v2f A  = *reinterpret_cast<const v2f*>(xr + k);
        v2f B0 = *reinterpret_cast<const v2f*>(w0 + k);
        v2f B1 = *reinterpret_cast<const v2f*>(w1 + k);
        v2f B2 = *reinterpret_cast<const v2f*>(w2 + k);
        v2f B3 = *reinterpret_cast<const v2f*>(w3 + k);
        acc0 = __builtin_amdgcn_wmma_f32_16x16x4_f32(false, A, false, B0,
                                                     (short)0, acc0, false, false);
        acc1 = __builtin_amdgcn_wmma_f32_16x16x4_f32(false, A, false, B1,
                                                     (short)0, acc1, false, false);
        acc2 = __builtin_amdgcn_wmma_f32_16x16x4_f32(false, A, false, B2,
                                                     (short)0, acc2, false, false);
        acc3 = __builtin_amdgcn_wmma_f32_16x16x4_f32(false, A, false, B3,
                                                     (short)0, acc3, false, false);
    }

    // ---------------- bias + stage to LDS, then coalesced b128 stores -------
    const float b0 = Bs[ 0 + lrow];
    const float b1 = Bs[16 + lrow];
    const float b2 = Bs[32 + lrow];
    const float b3 = Bs[48 + lrow];

    __syncthreads();                 // all waves done reading Xs -> reuse it
#pragma unroll
    for (int v = 0; v < 8; ++v) {
        // C/D layout: lanes 0-15 hold M=v, lanes 16-31 hold M=8+v; N = lane&15
        float* r = &Xs[(rbase + v + 8 * half) * LDSW + lrow];
        r[ 0] = acc0[v] + b0;
        r[16] = acc1[v] + b1;
        r[32] = acc2[v] + b2;
        r[48] = acc3[v] + b3;
    }
    __syncthreads();

    float4* og = reinterpret_cast<float4*>(out);
#pragma unroll
    for (int t = 0; t < 8; ++t) {    // 128 rows x 16 float4, 256B per row
        int j = tid + t * 256, row = j >> 4, col4 = j & 15;
        float4 val = *reinterpret_cast<const float4*>(&Xs[row * LDSW + col4 * 4]);
        og[(size_t)((m0 + row) * NBLK + n) * (DOUT / 4) + col4] = val;
    }
}

extern "C" void kernel_launch(void* const* d_in, const int* in_sizes, int n_in,
                              void* d_out, int out_size, void* d_ws, size_t ws_size,
                              hipStream_t stream)
{
    const float* x = (const float*)d_in[0];   // [1024, 1024, 64]
    const float* W = (const float*)d_in[1];   // [1024, 64, 64]
    const float* b = (const float*)d_in[2];   // [1024, 64]
    float* out = (float*)d_out;               // [1024, 1024, 64]

    dim3 grid(NBLK, BATCH / TILE_M);          // (1024, 8)
    block_linear_wmma_f32<<<grid, 256, 0, stream>>>(x, W, b, out);
}